// SequenceCompressor_65979287601682
// MI455X (gfx1250) — compile-verified
//
#include <hip/hip_runtime.h>
#include <math.h>

// SequenceCompressor for MI455X (gfx1250, wave32).
// B=8, S=4096, D=512 fixed by the reference's setup_inputs().
//
// Phase 1: probs = sigmoid(x @ W + b) via V_WMMA_F32_16X16X4_F32 (f32 WMMA
//          preserves reference numerics). 16 timesteps per wave, K looped 512/4.
// Phase 2: deterministic tree reduction -> mean -> adj scalar.
// Phase 3: per-batch keep flags + stable block scan -> compaction order,
//          counts, and the (B,1,S) mask output.
// Phase 4: bandwidth pass: indirect row gather with float4 (B128) moves.

typedef __attribute__((ext_vector_type(2))) float v2f;
typedef __attribute__((ext_vector_type(8))) float v8f;

#define NB   8
#define NS   4096
#define ND   512
#define NBS  (NB * NS)          // 32768 rows
#define EPSF 1e-7f

// ---------------------------------------------------------------------------
// Phase 1: per-row dot(x[r,:], W) + bias -> sigmoid, via f32 WMMA.
// One wave handles 16 consecutive rows (M dim). 8 waves / 256-thread block.
//
// f32 A (16x4, MxK) VGPR layout (ISA 7.12.2):
//   lanes 0-15 : M=lane,    v0=K0, v1=K1
//   lanes 16-31: M=lane-16, v0=K2, v1=K3
// f32 B (4x16, KxN) uses the same per-lane K mapping with N = lane&15; we
// replicate W across all N columns, so every lane in a 16-group loads the
// same W pair for its K slots.
// C/D (16x16 f32): VGPR j: lanes 0-15 -> M=j, lanes 16-31 -> M=j+8; any N
// column holds the full dot products, so lane 0 / lane 16 extract them.
// ---------------------------------------------------------------------------
__global__ __launch_bounds__(256) void sc_probs_wmma_kernel(
    const float* __restrict__ x, const float* __restrict__ W,
    const float* __restrict__ bias, float* __restrict__ probs) {
  const int lane   = threadIdx.x & 31;
  const int waveId = threadIdx.x >> 5;
  const int tile   = blockIdx.x * 8 + waveId;   // 16-row tile index
  const int r0     = tile * 16;
  if (r0 >= NBS) return;

  const int m      = lane & 15;
  const int laneHi = lane >> 4;                 // 0: K0/K1, 1: K2/K3
  const float* __restrict__ px = x + (size_t)(r0 + m) * ND;

  v8f acc = {};
  for (int kb = 0; kb < ND; kb += 4) {
    const int koff = kb + 2 * laneHi;
    v2f a, w;
    a[0] = px[koff];
    a[1] = px[koff + 1];
    w[0] = W[koff];
    w[1] = W[koff + 1];
    // D = A(16x4) * B(4x16) + C  ; accumulate over K=512
    acc = __builtin_amdgcn_wmma_f32_16x16x4_f32(
        /*neg_a=*/false, a, /*neg_b=*/false, w,
        /*c_mod=*/(short)0, acc, /*reuse_a=*/false, /*reuse_b=*/false);
  }

  // Column N=0 of C lives in lane 0 (M=0..7) and lane 16 (M=8..15).
  if (m == 0) {
    const float bi = bias[0];
    const int rowBase = r0 + laneHi * 8;
    #pragma unroll
    for (int j = 0; j < 8; ++j) {
      const float z = acc[j] + bi;
      probs[rowBase + j] = 1.0f / (1.0f + expf(-z));
    }
  }
}

// ---------------------------------------------------------------------------
// Phase 2: deterministic mean of probs -> adj = |cr + eps| / (mean + eps).
// Single block, fixed reduction order -> identical across graph replays.
// ---------------------------------------------------------------------------
__global__ __launch_bounds__(1024) void sc_adj_kernel(
    const float* __restrict__ probs, const float* __restrict__ cr,
    float* __restrict__ ws_adj) {
  __shared__ float sdata[1024];
  const int tid = threadIdx.x;
  float s = 0.0f;
  for (int i = tid; i < NBS; i += 1024) s += probs[i];
  sdata[tid] = s;
  __syncthreads();
  for (int off = 512; off > 0; off >>= 1) {
    if (tid < off) sdata[tid] += sdata[tid + off];
    __syncthreads();
  }
  if (tid == 0) {
    const float mean = sdata[0] / (float)NBS;
    ws_adj[0] = fabsf(cr[0] + EPSF) / (mean + EPSF);
  }
}

// ---------------------------------------------------------------------------
// Phase 3: one block per batch. keep flags -> stable exclusive scan ->
// compaction order + counts + mask output. 1024 threads x 4 items = S=4096.
// ---------------------------------------------------------------------------
__global__ __launch_bounds__(1024) void sc_scan_kernel(
    const float* __restrict__ probs, const float* __restrict__ u,
    const float* __restrict__ ws_adj, int* __restrict__ order,
    int* __restrict__ counts, float* __restrict__ mask_out) {
  __shared__ int sscan[1024];
  const int b    = blockIdx.x;
  const int tid  = threadIdx.x;
  const int base = b * NS;
  const float adj = ws_adj[0];

  int flags[4];
  int tsum = 0;
  #pragma unroll
  for (int i = 0; i < 4; ++i) {
    const int s = tid * 4 + i;
    float p = probs[base + s] * adj;
    p = fminf(p, 1.0f);
    p = fmaxf(p, 0.0f);
    if (s == 0) p = 1.0f;                 // first timestep always skip-prob 1
    const int skip = (u[base + s] < p) ? 1 : 0;
    flags[i] = 1 - skip;                  // keep flag
    tsum += flags[i];
  }

  sscan[tid] = tsum;
  __syncthreads();
  for (int off = 1; off < 1024; off <<= 1) {
    int v = 0;
    if (tid >= off) v = sscan[tid - off];
    __syncthreads();
    sscan[tid] += v;
    __syncthreads();
  }
  const int excl  = sscan[tid] - tsum;     // exclusive prefix over threads
  const int total = sscan[1023];

  // stable scatter of kept timestep indices
  int pos = excl;
  #pragma unroll
  for (int i = 0; i < 4; ++i) {
    const int s = tid * 4 + i;
    if (flags[i]) order[base + pos++] = s;
  }
  if (tid == 0) counts[b] = total;

  // mask (B,1,S): 1.0 for j < counts[b]
  #pragma unroll
  for (int i = 0; i < 4; ++i) {
    const int j = tid * 4 + i;
    mask_out[base + j] = (j < total) ? 1.0f : 0.0f;
  }
}

// ---------------------------------------------------------------------------
// Phase 4: gather rows. One block per output row; 128 lanes x float4 = 512
// floats (2 KB) per row, fully coalesced B128 traffic. Tail rows -> zeros.
// ---------------------------------------------------------------------------
__global__ __launch_bounds__(128) void sc_gather_kernel(
    const float* __restrict__ x, const int* __restrict__ order,
    const int* __restrict__ counts, float* __restrict__ out) {
  const int row = blockIdx.x;           // 0 .. NBS-1
  const int b   = row >> 12;            // row / NS
  const int j   = row & (NS - 1);       // row % NS
  const int tid = threadIdx.x;

  float4* __restrict__ dst = (float4*)(out + (size_t)row * ND);
  if (j < counts[b]) {
    const int src_s = order[b * NS + j];
    const float4* __restrict__ src =
        (const float4*)(x + ((size_t)b * NS + src_s) * ND);
    dst[tid] = src[tid];
  } else {
    dst[tid] = make_float4(0.0f, 0.0f, 0.0f, 0.0f);
  }
}

// ---------------------------------------------------------------------------
extern "C" void kernel_launch(void* const* d_in, const int* in_sizes, int n_in,
                              void* d_out, int out_size, void* d_ws, size_t ws_size,
                              hipStream_t stream) {
  (void)in_sizes; (void)n_in; (void)out_size; (void)ws_size;

  const float* x  = (const float*)d_in[0];   // (B,S,D)
  const float* W  = (const float*)d_in[1];   // (1,D)
  const float* bb = (const float*)d_in[2];   // (1,)
  const float* cr = (const float*)d_in[3];   // (1,)
  const float* u  = (const float*)d_in[4];   // (B,S,1)

  float* out_reduced = (float*)d_out;                          // B*S*D
  float* out_mask    = (float*)d_out + (size_t)NBS * ND;       // B*S

  // workspace layout
  float* ws_probs  = (float*)d_ws;                 // NBS floats
  int*   ws_order  = (int*)(ws_probs + NBS);       // NBS ints
  int*   ws_counts = (int*)(ws_order + NBS);       // NB ints
  float* ws_adj    = (float*)(ws_counts + NB);     // 1 float

  // Phase 1: 256 blocks x 8 waves x 16 rows = 32768 rows
  sc_probs_wmma_kernel<<<NBS / 128, 256, 0, stream>>>(x, W, bb, ws_probs);
  // Phase 2: scalar adj
  sc_adj_kernel<<<1, 1024, 0, stream>>>(ws_probs, cr, ws_adj);
  // Phase 3: per-batch compaction order + counts + mask
  sc_scan_kernel<<<NB, 1024, 0, stream>>>(ws_probs, u, ws_adj, ws_order,
                                          ws_counts, out_mask);
  // Phase 4: row gather (bandwidth-bound pass)
  sc_gather_kernel<<<NBS, 128, 0, stream>>>(x, ws_order, ws_counts, out_reduced);
}